// GSPolicyNetLSTM_32933809225890
// MI455X (gfx1250) — compile-verified
//
#include <hip/hip_runtime.h>
#include <math.h>

// Problem sizes (match reference)
#define IN_SHAPE 4096
#define H_S 250
#define H_R 100
#define MAX_LEN 30
#define VOCAB 40
#define OUT_CLS (512 * 512)

typedef float v8f_t __attribute__((ext_vector_type(8)));
typedef float v2f_t __attribute__((ext_vector_type(2)));

__device__ __forceinline__ float sigf(float x) { return 1.0f / (1.0f + expf(-x)); }

// ---------------------------------------------------------------------------
// WMMA matvec: out[0..M) = W(MxK, row-major) @ v   using V_WMMA_F32_16X16X4_F32
// The vector is broadcast across all 16 columns of B, so every column of the
// 16x16 D tile equals the matvec result; lanes 0 and 16 (which hold M=0..7 and
// M=8..15 in their 8 accumulator VGPRs) scatter the 16 rows to LDS.
// Must be called by all threads of a 256-thread block (8 waves, full EXEC for
// the wmma itself; tile loop bounds are wave-uniform).
// A-layout (ISA 7.12.2, 32-bit A 16x4): lanes 0-15 hold K={k0,k0+1},
// lanes 16-31 hold K={k0+2,k0+3}, M = lane&15.  B uses the same K split.
// ---------------------------------------------------------------------------
__device__ inline void matvec_wmma(const float* __restrict__ W, int M, int K,
                                   const float* __restrict__ v, float* out,
                                   bool accumulate) {
    const int lane = threadIdx.x & 31;
    const int wave = threadIdx.x >> 5;
    const int half = lane >> 4;   // 0: lanes 0-15, 1: lanes 16-31
    const int kOff = half << 1;   // K sub-offset 0 or 2
    const int mL = lane & 15;
    const int nTiles = (M + 15) >> 4;
    for (int tile = wave; tile < nTiles; tile += 8) {
        const int row0 = tile << 4;
        int row = row0 + mL;
        if (row >= M) row = M - 1;  // clamp (result discarded at store)
        const float* Wr = W + (size_t)row * (size_t)K;
        v8f_t acc = {};
        for (int k0 = 0; k0 < K; k0 += 4) {
            const int ka = k0 + kOff;
            const int kb = ka + 1;
            // zero B beyond K kills any clamped-A contribution
            const float bx = (ka < K) ? v[ka] : 0.0f;
            const float by = (kb < K) ? v[kb] : 0.0f;
            const int kaC = (ka < K) ? ka : (K - 1);
            const int kbC = (kb < K) ? kb : (K - 1);
            v2f_t A;
            A.x = Wr[kaC];
            A.y = Wr[kbC];
            v2f_t B;
            B.x = bx;
            B.y = by;
            acc = __builtin_amdgcn_wmma_f32_16x16x4_f32(
                /*neg_a=*/false, A, /*neg_b=*/false, B,
                /*c_mod=*/(short)0, acc, /*reuse_a=*/false, /*reuse_b=*/false);
        }
        if (mL == 0) {  // lanes 0 and 16 hold M=row0+half*8 .. +7 in acc[0..7]
            const int base = row0 + (half << 3);
#pragma unroll
            for (int j = 0; j < 8; ++j) {
                const int r = base + j;
                if (r < M) {
                    if (accumulate)
                        out[r] += acc[j];
                    else
                        out[r] = acc[j];
                }
            }
        }
    }
}

// ---------------------------------------------------------------------------
// K1: h0 = relu(W_s1 @ x + b_s1)   (one block per output row)
// ---------------------------------------------------------------------------
__global__ __launch_bounds__(256) void k_h0(const float* __restrict__ x,
                                            const float* __restrict__ W,
                                            const float* __restrict__ b,
                                            float* __restrict__ out) {
    __shared__ float red[8];
    const int t = threadIdx.x, lane = t & 31, wave = t >> 5;
    const float* Wr = W + (size_t)blockIdx.x * IN_SHAPE;
    float acc = 0.0f;
    for (int k = t; k < IN_SHAPE; k += 256) acc += Wr[k] * x[k];
    for (int o = 16; o > 0; o >>= 1) acc += __shfl_xor(acc, o, 32);
    if (lane == 0) red[wave] = acc;
    __syncthreads();
    if (t == 0) {
        float s = 0.0f;
        for (int w = 0; w < 8; ++w) s += red[w];
        s += b[blockIdx.x];
        out[blockIdx.x] = fmaxf(s, 0.0f);
    }
}

// ---------------------------------------------------------------------------
// K2: the full sequential part (sender 30 steps + receiver 30 steps), one
// 256-thread workgroup, all state in LDS, all matvecs through WMMA.
// ---------------------------------------------------------------------------
__global__ __launch_bounds__(256) void k_seq(
    const float* __restrict__ gumbel, const float* __restrict__ W_ih1,
    const float* __restrict__ W_hh1, const float* __restrict__ b_ih1,
    const float* __restrict__ b_hh1, const float* __restrict__ W_p,
    const float* __restrict__ b_p, const float* __restrict__ W_ih2,
    const float* __restrict__ W_hh2, const float* __restrict__ b_ih2,
    const float* __restrict__ b_hh2, const float* __restrict__ h0_in,
    float* __restrict__ hR_out) {
    __shared__ float h[H_S], c[H_S], nh[H_S], nc[H_S];
    __shared__ float sym[VOCAB], soft[VOCAB], logits[VOCAB];
    __shared__ float gates[4 * H_S];
    __shared__ float msg[MAX_LEN][VOCAB];
    __shared__ int valid[MAX_LEN];
    __shared__ float hR[H_R], cR[H_R], gatesR[4 * H_R];
    __shared__ int s_doneIn, s_idx, s_done;

    const int t = threadIdx.x;
    if (t < H_S) {
        h[t] = h0_in[t];
        c[t] = 0.0f;
    }
    if (t < VOCAB) sym[t] = 0.0f;  // SOS
    if (t == 0) s_done = 0;
    __syncthreads();

    // ---- sender ----
    for (int step = 0; step < MAX_LEN; ++step) {
        matvec_wmma(W_ih1, 4 * H_S, VOCAB, sym, gates, false);
        matvec_wmma(W_hh1, 4 * H_S, H_S, h, gates, true);
        __syncthreads();
        if (t < H_S) {
            const float gi = gates[t] + b_ih1[t] + b_hh1[t];
            const float gf = gates[H_S + t] + b_ih1[H_S + t] + b_hh1[H_S + t];
            const float gg = gates[2 * H_S + t] + b_ih1[2 * H_S + t] + b_hh1[2 * H_S + t];
            const float go = gates[3 * H_S + t] + b_ih1[3 * H_S + t] + b_hh1[3 * H_S + t];
            const float cn = sigf(gf) * c[t] + sigf(gi) * tanhf(gg);
            nc[t] = cn;
            nh[t] = sigf(go) * tanhf(cn);
        }
        __syncthreads();
        matvec_wmma(W_p, VOCAB, H_S, nh, logits, false);
        __syncthreads();
        if (t == 0) {
            const float* g = gumbel + step * VOCAB;
            float z[VOCAB];
            float m = -INFINITY;
            for (int v = 0; v < VOCAB; ++v) {
                z[v] = logits[v] + b_p[v] + g[v];
                m = fmaxf(m, z[v]);
            }
            float s = 0.0f;
            for (int v = 0; v < VOCAB; ++v) {
                z[v] = expf(z[v] - m);
                s += z[v];
            }
            const float inv = 1.0f / s;
            int idx = 0;
            float best = -1.0f;
            for (int v = 0; v < VOCAB; ++v) {
                const float sv = z[v] * inv;
                soft[v] = sv;
                if (sv > best) {  // first-max wins, like jnp.argmax
                    best = sv;
                    idx = v;
                }
            }
            s_idx = idx;
            s_doneIn = s_done;
            const int last = (step == MAX_LEN - 1) ? 1 : 0;
            s_done = s_done | ((idx == VOCAB - 1) ? 1 : 0) | last;
        }
        __syncthreads();
        const int done = s_doneIn;
        const int last = (step == MAX_LEN - 1);
        if (t < VOCAB) {
            const float hard = (t == s_idx) ? 1.0f : 0.0f;
            const float st = hard + soft[t] - soft[t];  // straight-through
            const float eos = (t == VOCAB - 1) ? 1.0f : 0.0f;
            float emit = (last && !done) ? eos : st;
            emit = done ? eos : emit;
            msg[step][t] = emit;
            if (!done) sym[t] = emit;
        }
        if (t == 0) valid[step] = !done;
        if (t < H_S && !done) {
            h[t] = nh[t];
            c[t] = nc[t];
        }
        __syncthreads();
    }

    // ---- receiver ----
    if (t < H_R) {
        hR[t] = 0.0f;
        cR[t] = 0.0f;
    }
    __syncthreads();
    for (int step = 0; step < MAX_LEN; ++step) {
        matvec_wmma(W_ih2, 4 * H_R, VOCAB, msg[step], gatesR, false);
        matvec_wmma(W_hh2, 4 * H_R, H_R, hR, gatesR, true);
        __syncthreads();
        if (t < H_R && valid[step]) {
            const float gi = gatesR[t] + b_ih2[t] + b_hh2[t];
            const float gf = gatesR[H_R + t] + b_ih2[H_R + t] + b_hh2[H_R + t];
            const float gg = gatesR[2 * H_R + t] + b_ih2[2 * H_R + t] + b_hh2[2 * H_R + t];
            const float go = gatesR[3 * H_R + t] + b_ih2[3 * H_R + t] + b_hh2[3 * H_R + t];
            const float cn = sigf(gf) * cR[t] + sigf(gi) * tanhf(gg);
            cR[t] = cn;
            hR[t] = sigf(go) * tanhf(cn);
        }
        __syncthreads();
    }
    if (t < H_R) hR_out[t] = hR[t];
}

// ---------------------------------------------------------------------------
// K3: logits_r = W_r @ hR + b_r  (wave-per-row, lane-coalesced 128B loads),
// plus per-block max. 64 rows per block -> grid = OUT_CLS/64.
// ---------------------------------------------------------------------------
__global__ __launch_bounds__(256) void k_bigmatvec(const float* __restrict__ W_r,
                                                   const float* __restrict__ b_r,
                                                   const float* __restrict__ hRg,
                                                   float* __restrict__ out_logits,
                                                   float* __restrict__ blockMax) {
    __shared__ float s_h[128];
    __shared__ float s_wmax[8];
    const int t = threadIdx.x, lane = t & 31, wave = t >> 5;
    if (t < 128) s_h[t] = (t < H_R) ? hRg[t] : 0.0f;
    __syncthreads();
    const float h0 = s_h[lane];
    const float h1 = s_h[lane + 32];
    const float h2 = s_h[lane + 64];
    const float h3 = s_h[lane + 96];  // zero for lane>=4
    float wmax = -INFINITY;
    const int rowBase = blockIdx.x * 64;
#pragma unroll
    for (int i = 0; i < 8; ++i) {
        const int row = rowBase + wave + i * 8;
        const float* Wr = W_r + (size_t)row * H_R;
        if (i < 7) __builtin_prefetch(W_r + (size_t)(row + 8) * H_R, 0, 1);
        float acc = Wr[lane] * h0 + Wr[lane + 32] * h1 + Wr[lane + 64] * h2;
        const float w3 = (lane < 4) ? Wr[96 + lane] : 0.0f;  // guard OOB on last row
        acc += w3 * h3;
        for (int o = 16; o > 0; o >>= 1) acc += __shfl_xor(acc, o, 32);
        if (lane == 0) {
            const float v = acc + b_r[row];
            out_logits[row] = v;
            wmax = fmaxf(wmax, v);
        }
    }
    if (lane == 0) s_wmax[wave] = wmax;
    __syncthreads();
    if (t == 0) {
        float m = -INFINITY;
        for (int w = 0; w < 8; ++w) m = fmaxf(m, s_wmax[w]);
        blockMax[blockIdx.x] = m;
    }
}

// ---------------------------------------------------------------------------
// K4/K6: deterministic single-block reductions
// ---------------------------------------------------------------------------
__global__ __launch_bounds__(256) void k_reduce_max(const float* __restrict__ in,
                                                    int n, float* __restrict__ out) {
    __shared__ float red[256];
    float m = -INFINITY;
    for (int i = threadIdx.x; i < n; i += 256) m = fmaxf(m, in[i]);
    red[threadIdx.x] = m;
    __syncthreads();
    for (int s = 128; s > 0; s >>= 1) {
        if (threadIdx.x < s) red[threadIdx.x] = fmaxf(red[threadIdx.x], red[threadIdx.x + s]);
        __syncthreads();
    }
    if (threadIdx.x == 0) out[0] = red[0];
}

__global__ __launch_bounds__(256) void k_reduce_sum(const float* __restrict__ in,
                                                    int n, float* __restrict__ out) {
    __shared__ float red[256];
    float a = 0.0f;
    for (int i = threadIdx.x; i < n; i += 256) a += in[i];
    red[threadIdx.x] = a;
    __syncthreads();
    for (int s = 128; s > 0; s >>= 1) {
        if (threadIdx.x < s) red[threadIdx.x] += red[threadIdx.x + s];
        __syncthreads();
    }
    if (threadIdx.x == 0) out[0] = red[0];
}

// K5: in-place exp(x - gmax) + per-block sums (256 elems per block)
__global__ __launch_bounds__(256) void k_exp(float* __restrict__ data,
                                             const float* __restrict__ gmax,
                                             float* __restrict__ blockSum) {
    __shared__ float red[256];
    const int i = blockIdx.x * 256 + threadIdx.x;
    const float e = expf(data[i] - gmax[0]);
    data[i] = e;
    red[threadIdx.x] = e;
    __syncthreads();
    for (int s = 128; s > 0; s >>= 1) {
        if (threadIdx.x < s) red[threadIdx.x] += red[threadIdx.x + s];
        __syncthreads();
    }
    if (threadIdx.x == 0) blockSum[blockIdx.x] = red[0];
}

// K7: normalize
__global__ __launch_bounds__(256) void k_norm(float* __restrict__ data,
                                              const float* __restrict__ gsum) {
    const int i = blockIdx.x * 256 + threadIdx.x;
    data[i] = data[i] / gsum[0];
}

// ---------------------------------------------------------------------------
extern "C" void kernel_launch(void* const* d_in, const int* in_sizes, int n_in,
                              void* d_out, int out_size, void* d_ws, size_t ws_size,
                              hipStream_t stream) {
    const float* x = (const float*)d_in[0];
    const float* gumbel = (const float*)d_in[1];
    const float* W_s1 = (const float*)d_in[2];
    const float* b_s1 = (const float*)d_in[3];
    const float* W_ih1 = (const float*)d_in[4];
    const float* W_hh1 = (const float*)d_in[5];
    const float* b_ih1 = (const float*)d_in[6];
    const float* b_hh1 = (const float*)d_in[7];
    const float* W_p = (const float*)d_in[8];
    const float* b_p = (const float*)d_in[9];
    const float* W_ih2 = (const float*)d_in[10];
    const float* W_hh2 = (const float*)d_in[11];
    const float* b_ih2 = (const float*)d_in[12];
    const float* b_hh2 = (const float*)d_in[13];
    const float* W_r = (const float*)d_in[14];
    const float* b_r = (const float*)d_in[15];

    float* out = (float*)d_out;  // OUT_CLS floats
    float* ws = (float*)d_ws;
    float* h0 = ws + 0;                   // 250
    float* hR = ws + 256;                 // 100
    float* bmax = ws + 512;               // 4096
    float* bsum = ws + 512 + 4096;        // 1024
    float* gmax = ws + 512 + 4096 + 1024; // 1
    float* gsum = gmax + 1;               // 1

    k_h0<<<H_S, 256, 0, stream>>>(x, W_s1, b_s1, h0);
    k_seq<<<1, 256, 0, stream>>>(gumbel, W_ih1, W_hh1, b_ih1, b_hh1, W_p, b_p,
                                 W_ih2, W_hh2, b_ih2, b_hh2, h0, hR);
    k_bigmatvec<<<OUT_CLS / 64, 256, 0, stream>>>(W_r, b_r, hR, out, bmax);
    k_reduce_max<<<1, 256, 0, stream>>>(bmax, OUT_CLS / 64, gmax);
    k_exp<<<OUT_CLS / 256, 256, 0, stream>>>(out, gmax, bsum);
    k_reduce_sum<<<1, 256, 0, stream>>>(bsum, OUT_CLS / 256, gsum);
    k_norm<<<OUT_CLS / 256, 256, 0, stream>>>(out, gsum);
}